// AFNO_35983236006656
// MI455X (gfx1250) — compile-verified
//
#include <hip/hip_runtime.h>
#include <math.h>

// AFNO on MI455X (gfx1250, wave32). All dense math via V_WMMA_F32_16X16X4_F32
// (exact f32 matrix cores). DFTs expressed as GEMMs vs precomputed cos/sin
// matrices. HBM-bound problem: each streamed operand is read from HBM exactly
// once (full-row-block waves + LDS channel staging in coldft), ~2.05 GB total
// traffic @ 23.3 TB/s. All inner-loop loads are unconditional (padded
// operands) so no exec-mask branches around WMMA. The coldft LDS staging uses
// gfx1250 GLOBAL_LOAD_ASYNC_TO_LDS_B128 (ASYNCcnt-tracked, bypasses VGPRs)
// when the toolchain exposes the builtin.
//
// Spectral layout: S[bc][h][132] : cols 0..64 = Re, 65..129 = Im, 130..131 = 0.

typedef __attribute__((ext_vector_type(2))) float v2f;
typedef __attribute__((ext_vector_type(8))) float v8f;
typedef __attribute__((ext_vector_type(4))) int   v4i;

#define NHEADS 32
#define SHD    24
#define HID    96
#define BB     4
#define CC     768
#define HH     128
#define WW     128
#define WF     65
#define RL     132                 // padded spectral row length
#define CHF    (HH * RL)           // floats per channel (16896)
#define LMBDA  0.01f

#if __has_builtin(__builtin_amdgcn_global_load_async_to_lds_b128)
#define USE_ASYNC_LDS 1
#else
#define USE_ASYNC_LDS 0
#endif

__device__ __forceinline__ v8f wmma4(v2f a, v2f b, v8f c) {
  return __builtin_amdgcn_wmma_f32_16x16x4_f32(false, a, false, b, (short)0, c,
                                               false, false);
}

// ---------------------------------------------------------------------------
// Init: DFT matrices (K-paired for single-b64 B loads) + packed MLP weights.
//  Fwp  [64 pairs][144] v2f : {Fw[2p][j], Fw[2p+1][j]}, Fw[w][j<65]=cos/√128,
//                             Fw[w][65..129]=-sin/√128, j>=130 -> 0
//  iFwp [66 pairs][128] v2f : Hermitian-folded inverse rows, rows>=130 -> 0
//  Ch/Sh [128][128]         : column DFT cos/sin /√128 (A-side, row-contig)
//  W1p  [2*32][12 pairs][96] v2f, W2p [2*32][48 pairs][24] v2f
// ---------------------------------------------------------------------------
__global__ void afno_init(const float* __restrict__ w1,
                          const float* __restrict__ w2,
                          v2f* __restrict__ Fwp, v2f* __restrict__ iFwp,
                          float* __restrict__ Ch, float* __restrict__ Sh,
                          v2f* __restrict__ W1p, v2f* __restrict__ W2p) {
  const int idx = blockIdx.x * blockDim.x + threadIdx.x;
  const float inv = 0.08838834764831845f;          // 1/sqrt(128)
  const float step = 6.283185307179586f / 128.0f;  // 2*pi/128

  if (idx < 64 * 144) {                            // Fwp
    int p = idx / 144, j = idx % 144;
    float e0 = 0.0f, e1 = 0.0f;
    if (j < 130) {
      int k = (j < WF) ? j : j - WF;
      int m0 = (k * (2 * p)) & 127;
      int m1 = (k * (2 * p + 1)) & 127;
      if (j < WF) {
        e0 = cosf(step * (float)m0) * inv;
        e1 = cosf(step * (float)m1) * inv;
      } else {
        e0 = -sinf(step * (float)m0) * inv;
        e1 = -sinf(step * (float)m1) * inv;
      }
    }
    Fwp[idx] = (v2f){e0, e1};
  }
  if (idx < 66 * 128) {                            // iFwp
    int p = idx / 128, w = idx % 128;
    float e[2];
    for (int q = 0; q < 2; ++q) {
      int k = 2 * p + q;
      float v = 0.0f;
      if (k < 130) {
        int kk = (k < WF) ? k : k - WF;
        float alpha = (kk == 0 || kk == 64) ? 1.0f : 2.0f;
        int m = (kk * w) & 127;
        v = ((k < WF) ? cosf(step * (float)m) : -sinf(step * (float)m)) *
            alpha * inv;
      }
      e[q] = v;
    }
    iFwp[idx] = (v2f){e[0], e[1]};
  }
  if (idx < 128 * 128) {                           // Ch / Sh
    int ky = idx / 128, h = idx % 128;
    int m = (ky * h) & 127;
    Ch[idx] = cosf(step * (float)m) * inv;
    Sh[idx] = sinf(step * (float)m) * inv;
  }
  if (idx < 2 * NHEADS * 12 * HID) {               // W1p : pairs over sh
    int c = idx % HID;
    int q = idx / HID;                             // tn*12 + p
    int tn = q / 12, p = q % 12;
    const float* src = w1 + (size_t)tn * SHD * HID;
    W1p[idx] = (v2f){src[(2 * p) * HID + c], src[(2 * p + 1) * HID + c]};
  }
  if (idx < 2 * NHEADS * 48 * SHD) {               // W2p : pairs over hidden
    int s = idx % SHD;
    int q = idx / SHD;                             // tn*48 + p
    int tn = q / 48, p = q % 48;
    const float* src = w2 + (size_t)tn * HID * SHD;
    W2p[idx] = (v2f){src[(2 * p) * SHD + s], src[(2 * p + 1) * SHD + s]};
  }
}

// ---------------------------------------------------------------------------
// Stage A: row rfft. One wave per 16-row M-tile computes ALL 9 N-tiles
// (x read from HBM exactly once). K=128, unconditional loads.
// ---------------------------------------------------------------------------
__global__ void afno_rowfft(const float* __restrict__ x,
                            const v2f* __restrict__ Fwp,
                            float* __restrict__ T) {
  const int lane = threadIdx.x;
  const int m0 = (int)blockIdx.x * 16;
  const int r = lane & 15, col = lane & 15;
  const int ko = (lane < 16) ? 0 : 2;

  const float* arow = x + (size_t)(m0 + r) * WW;
  v8f acc[9];
#pragma unroll
  for (int t = 0; t < 9; ++t) acc[t] = (v8f){};

  for (int k = 0; k < WW; k += 4) {
    const v2f a = *(const v2f*)(arow + k + ko);
    const int pr = (k + ko) >> 1;
#pragma unroll
    for (int t = 0; t < 9; ++t) {
      v2f b = Fwp[pr * 144 + t * 16 + col];
      acc[t] = wmma4(a, b, acc[t]);
    }
  }
#pragma unroll
  for (int t = 0; t < 9; ++t) {
    const int j = t * 16 + col;
    if (j < RL) {                      // masked store only (incl. zero pads)
#pragma unroll
      for (int v = 0; v < 8; ++v) {
        const int m = m0 + v + 8 * (lane >> 4);
        T[(size_t)m * RL + j] = acc[t][v];
      }
    }
  }
}

// ---------------------------------------------------------------------------
// Stage B/D: column DFT per channel, sig=+1 fwd / -1 inv.
// 256-thread block stages the whole channel (132x128 f32 = 67.6 KB) into LDS
// once (async direct-to-LDS when available); 8 waves each own one ky-tile and
// sweep the 5 kx-tiles.
//   Zr = C@Yr + sig*S@Yi ; Zi = C@Yi - sig*S@Yr
// ---------------------------------------------------------------------------
__global__ __launch_bounds__(256) void afno_coldft(
    const float* __restrict__ Tin, float* __restrict__ Tout,
    const float* __restrict__ Ch, const float* __restrict__ Sh, float sig) {
  __shared__ float Y[CHF];

  const int bc = (int)blockIdx.x;
  const int tid = threadIdx.x;
#if USE_ASYNC_LDS
  {
    // GLOBAL_LOAD_ASYNC_TO_LDS_B128: memory -> LDS without VGPR staging,
    // tracked by ASYNCcnt. Builtin takes int4-typed pointers.
    __attribute__((address_space(1))) v4i* g =
        (__attribute__((address_space(1))) v4i*)(Tin + (size_t)bc * CHF);
    __attribute__((address_space(3))) v4i* l =
        (__attribute__((address_space(3))) v4i*)Y;
    for (int i = tid; i < CHF / 4; i += 256)
      __builtin_amdgcn_global_load_async_to_lds_b128(g + i, l + i, 0, 0);
#if __has_builtin(__builtin_amdgcn_s_wait_asynccnt)
    __builtin_amdgcn_s_wait_asynccnt(0);
#else
    asm volatile("s_wait_asynccnt 0x0" ::: "memory");
#endif
  }
#else
  {
    const float4* src = (const float4*)(Tin + (size_t)bc * CHF);
    float4* dst = (float4*)Y;
    for (int i = tid; i < CHF / 4; i += 256) dst[i] = src[i];
  }
#endif
  __syncthreads();

  const int lane = tid & 31;
  const int ky0 = (tid >> 5) * 16;
  const int r = lane & 15, col = lane & 15;
  const int ko = (lane < 16) ? 0 : 2;
  float* Z = Tout + (size_t)bc * CHF;

  for (int xt = 0; xt < 5; ++xt) {
    const int kx = xt * 16 + col;
    const bool cok = (kx < WF);
    const int kxc = cok ? kx : 0;                  // index clamp, no branch

    v8f crr = {}, cri = {}, csr = {}, csi = {};
    for (int h = 0; h < HH; h += 4) {
      const v2f aC = *(const v2f*)(Ch + (ky0 + r) * HH + h + ko);
      const v2f aS = *(const v2f*)(Sh + (ky0 + r) * HH + h + ko);
      v2f bR, bI;
      bR.x = Y[(h + ko) * RL + kxc];
      bR.y = Y[(h + ko + 1) * RL + kxc];
      bI.x = Y[(h + ko) * RL + WF + kxc];
      bI.y = Y[(h + ko + 1) * RL + WF + kxc];
      crr = wmma4(aC, bR, crr);
      cri = wmma4(aC, bI, cri);
      csr = wmma4(aS, bR, csr);
      csi = wmma4(aS, bI, csi);
    }
    if (cok) {
#pragma unroll
      for (int v = 0; v < 8; ++v) {
        const int ky = ky0 + v + 8 * (lane >> 4);
        Z[(size_t)ky * RL + kx]      = crr[v] + sig * csi[v];
        Z[(size_t)ky * RL + WF + kx] = cri[v] - sig * csr[v];
      }
    }
  }
}

// ---------------------------------------------------------------------------
// Stage C: per-head complex MLP (24 -> 96 relu -> 24, softshrink), in place.
//  t0=xr@W1[0], t1=xi@W1[1]; x1r=relu(t0-t1+b1[0]); x1i=relu(t0+t1+b1[1])
//  u0=x1r@W2[0], u1=x1i@W2[1]; x2=softshrink(u0∓u1+b2)
// ---------------------------------------------------------------------------
__global__ void afno_mlp(float* __restrict__ T, const v2f* __restrict__ W1p,
                         const float* __restrict__ b1,
                         const v2f* __restrict__ W2p,
                         const float* __restrict__ b2) {
  __shared__ float x1r[16][HID + 4];
  __shared__ float x1i[16][HID + 4];

  const int lane = threadIdx.x;
  const int n = blockIdx.y;
  const int p0 = (int)blockIdx.x * 16;
  const int r = lane & 15, col = lane & 15;
  const int ko = (lane < 16) ? 0 : 2;

  const int p = p0 + r;
  const int b = p / (HH * WF);
  const int rem = p % (HH * WF);
  const int ky = rem / WF;
  const int kx = rem % WF;
  const size_t cstr = (size_t)CHF;                 // channel stride
  const size_t base = ((size_t)(b * CC + n * SHD) * HH + ky) * RL + kx;

  const v2f* W10 = W1p + (size_t)(0 * NHEADS + n) * 12 * HID;
  const v2f* W11 = W1p + (size_t)(1 * NHEADS + n) * 12 * HID;

  for (int nt = 0; nt < 6; ++nt) {
    const int c0 = nt * 16;
    v8f t0 = {}, t1 = {};
    for (int s = 0; s < SHD; s += 4) {
      v2f ar, ai;
      ar.x = T[base + (size_t)(s + ko) * cstr];
      ar.y = T[base + (size_t)(s + ko + 1) * cstr];
      ai.x = T[base + (size_t)(s + ko) * cstr + WF];
      ai.y = T[base + (size_t)(s + ko + 1) * cstr + WF];
      const v2f bw0 = W10[((s + ko) >> 1) * HID + c0 + col];
      const v2f bw1 = W11[((s + ko) >> 1) * HID + c0 + col];
      t0 = wmma4(ar, bw0, t0);
      t1 = wmma4(ai, bw1, t1);
    }
    const float bias0 = b1[(0 * NHEADS + n) * HID + c0 + col];
    const float bias1 = b1[(1 * NHEADS + n) * HID + c0 + col];
#pragma unroll
    for (int v = 0; v < 8; ++v) {
      const int row = v + 8 * (lane >> 4);
      float vr = t0[v] - t1[v] + bias0;
      float vi = t0[v] + t1[v] + bias1;
      x1r[row][c0 + col] = vr > 0.0f ? vr : 0.0f;
      x1i[row][c0 + col] = vi > 0.0f ? vi : 0.0f;
    }
  }
  __syncthreads();

  const v2f* W20 = W2p + (size_t)(0 * NHEADS + n) * 48 * SHD;
  const v2f* W21 = W2p + (size_t)(1 * NHEADS + n) * 48 * SHD;
  for (int st = 0; st < 2; ++st) {
    const int sc = st * 16 + col;
    const int scc = (sc < SHD) ? sc : 0;           // index clamp
    v8f u0 = {}, u1 = {};
    for (int k = 0; k < HID; k += 4) {
      v2f ar, ai;
      ar.x = x1r[r][k + ko];
      ar.y = x1r[r][k + ko + 1];
      ai.x = x1i[r][k + ko];
      ai.y = x1i[r][k + ko + 1];
      const v2f bw0 = W20[((k + ko) >> 1) * SHD + scc];
      const v2f bw1 = W21[((k + ko) >> 1) * SHD + scc];
      u0 = wmma4(ar, bw0, u0);
      u1 = wmma4(ai, bw1, u1);
    }
    if (sc < SHD) {
      const float bb0 = b2[(0 * NHEADS + n) * SHD + sc];
      const float bb1 = b2[(1 * NHEADS + n) * SHD + sc];
#pragma unroll
      for (int v = 0; v < 8; ++v) {
        const int row = v + 8 * (lane >> 4);
        const int pp = p0 + row;
        const int bb = pp / (HH * WF);
        const int rr2 = pp % (HH * WF);
        const int kyy = rr2 / WF;
        const int kxx = rr2 % WF;
        const size_t ob =
            ((size_t)(bb * CC + n * SHD + sc) * HH + kyy) * RL + kxx;
        float vr = u0[v] - u1[v] + bb0;
        float vi = u0[v] + u1[v] + bb1;
        vr = (vr > LMBDA) ? vr - LMBDA : ((vr < -LMBDA) ? vr + LMBDA : 0.0f);
        vi = (vi > LMBDA) ? vi - LMBDA : ((vi < -LMBDA) ? vi + LMBDA : 0.0f);
        T[ob] = vr;
        T[ob + WF] = vi;
      }
    }
  }
}

// ---------------------------------------------------------------------------
// Stage E: inverse row rfft. One wave per 16-row M-tile computes ALL 8
// N-tiles. A = contiguous padded spectral row (132 f32, pads zero) ->
// unconditional b64 loads; K=132.
// ---------------------------------------------------------------------------
__global__ void afno_irowfft(const float* __restrict__ T,
                             const v2f* __restrict__ iFwp,
                             float* __restrict__ out) {
  const int lane = threadIdx.x;
  const int m0 = (int)blockIdx.x * 16;
  const int r = lane & 15, col = lane & 15;
  const int ko = (lane < 16) ? 0 : 2;

  const float* arow = T + (size_t)(m0 + r) * RL;
  v8f acc[8];
#pragma unroll
  for (int t = 0; t < 8; ++t) acc[t] = (v8f){};

  for (int k = 0; k < RL; k += 4) {
    const v2f a = *(const v2f*)(arow + k + ko);
    const int pr = (k + ko) >> 1;
#pragma unroll
    for (int t = 0; t < 8; ++t) {
      v2f b = iFwp[pr * 128 + t * 16 + col];
      acc[t] = wmma4(a, b, acc[t]);
    }
  }
#pragma unroll
  for (int t = 0; t < 8; ++t) {
#pragma unroll
    for (int v = 0; v < 8; ++v) {
      const int row = m0 + v + 8 * (lane >> 4);
      out[(size_t)row * WW + t * 16 + col] = acc[t][v];
    }
  }
}

// ---------------------------------------------------------------------------
extern "C" void kernel_launch(void* const* d_in, const int* in_sizes, int n_in,
                              void* d_out, int out_size, void* d_ws,
                              size_t ws_size, hipStream_t stream) {
  const float* x  = (const float*)d_in[0];
  const float* w1 = (const float*)d_in[1];
  const float* b1 = (const float*)d_in[2];
  const float* w2 = (const float*)d_in[3];
  const float* b2 = (const float*)d_in[4];
  float* out = (float*)d_out;

  // Workspace layout (floats)
  float* ws = (float*)d_ws;
  v2f*   Fwp  = (v2f*)ws;                          // 64*144 pairs  (18432 f)
  v2f*   iFwp = (v2f*)(ws + 18432);                // 66*128 pairs  (16896 f)
  float* Ch   = ws + 18432 + 16896;                // 16384 f
  float* Sh   = Ch + 16384;                        // 16384 f
  v2f*   W1p  = (v2f*)(Sh + 16384);                // 73728 pairs  (147456 f)
  v2f*   W2p  = (v2f*)(Sh + 16384 + 147456);       // 73728 pairs  (147456 f)
  float* T1   = Sh + 16384 + 2 * 147456;           // 3072*16896 f
  const size_t specN = (size_t)BB * CC * CHF;
  float* T2   = T1 + specN;

  // 1) DFT matrices + weight packing (73728 threads covers every table)
  afno_init<<<(73728 + 255) / 256, 256, 0, stream>>>(w1, w2, Fwp, iFwp, Ch, Sh,
                                                     W1p, W2p);
  // 2) Row rfft: x -> T1
  afno_rowfft<<<(BB * CC * HH) / 16, 32, 0, stream>>>(x, Fwp, T1);
  // 3) Column DFT forward: T1 -> T2
  afno_coldft<<<BB * CC, 256, 0, stream>>>(T1, T2, Ch, Sh, 1.0f);
  // 4) Per-head complex MLP + softshrink, in place on T2
  {
    dim3 grid((BB * HH * WF) / 16, NHEADS);
    afno_mlp<<<grid, 32, 0, stream>>>(T2, W1p, b1, W2p, b2);
  }
  // 5) Column DFT inverse: T2 -> T1
  afno_coldft<<<BB * CC, 256, 0, stream>>>(T2, T1, Ch, Sh, -1.0f);
  // 6) Inverse row rfft: T1 -> out
  afno_irowfft<<<(BB * CC * HH) / 16, 32, 0, stream>>>(T1, iFwp, out);
}